// Implicit_Transformer_Up_20478404067514
// MI455X (gfx1250) — compile-verified
//
#include <hip/hip_runtime.h>
#include <math.h>

// CDNA5 / gfx1250. wave32.
typedef __attribute__((ext_vector_type(2))) float v2f;
typedef __attribute__((ext_vector_type(8))) float v8f;

#define D 256  // MLP hidden dim

__device__ __forceinline__ float gelu_exact(float x) {
    // jax.nn.gelu(approximate=False): x * 0.5 * (1 + erf(x / sqrt(2)))
    return 0.5f * x * (1.0f + erff(x * 0.70710678118654752440f));
}

// ---------------------------------------------------------------------------
// Kernel 1: the entire MLP collapses to ONE forward pass on the constant
// input [0,0,1,1] (rel==0 exactly, scale*hw==1 exactly). 16 waves, one
// 16-column output tile per wave, K-accumulation via V_WMMA_F32_16X16X4_F32.
// All 16 M-rows of the A matrix carry the same activation vector, so every
// row of the D tile equals the layer output; row 0 lives in acc[0] at
// lane N = lane&15 (ISA 7.12.2 f32 C/D layout).
// ---------------------------------------------------------------------------
__global__ __launch_bounds__(512) void mlp_weight_kernel(
    const float* __restrict__ W1, const float* __restrict__ b1,
    const float* __restrict__ W2, const float* __restrict__ b2,
    const float* __restrict__ W3, const float* __restrict__ b3,
    const float* __restrict__ W4, const float* __restrict__ b4,
    const float* __restrict__ W5, const float* __restrict__ b5,
    float* __restrict__ w12_out)
{
    __shared__ float bufA[D];
    __shared__ float bufB[D];

    const int tid  = threadIdx.x;
    const int wave = tid >> 5;      // 0..15 -> output column tile
    const int lane = tid & 31;
    const int n    = lane & 15;     // N within tile / M row id
    const int hi   = (lane >> 4) & 1; // half-wave selects K pair (0,1) vs (2,3)

    // Layer 1: input is exactly [0,0,1,1] -> pre-act[j] = W1[j,2] + W1[j,3] + b1[j]
    if (tid < D) {
        float v = W1[tid * 4 + 2] + W1[tid * 4 + 3] + b1[tid];
        bufA[tid] = gelu_exact(v);
    }
    __syncthreads();

    const float* Ws[3] = {W2, W3, W4};
    const float* bs[3] = {b2, b3, b4};
    float* cur = bufA;
    float* nxt = bufB;

    for (int L = 0; L < 3; ++L) {
        const float* Wm = Ws[L];
        const float* bb = bs[L];
        const int tile  = wave;
        // y[j] = sum_k x[k] * W[j,k]  (x @ W.T). B[k][n] = W[tile*16+n][k0+k]:
        // row of W is contiguous, so each lane loads a float pair per step.
        const float* wrow = Wm + (size_t)(tile * 16 + n) * D;

        v8f acc = {};
        for (int k0 = 0; k0 < D; k0 += 4) {
            v2f a, bmat;
            // A 16x4 f32 layout: lanes 0-15 -> K=0(V0),1(V1); lanes 16-31 -> K=2,3.
            a[0]    = cur[k0 + 2 * hi];
            a[1]    = cur[k0 + 2 * hi + 1];
            // B 4x16 f32 layout: low half K=0(V0),1(V1); high half K=2(V0),3(V1); N=lane&15.
            bmat[0] = wrow[k0 + 2 * hi];
            bmat[1] = wrow[k0 + 2 * hi + 1];
            acc = __builtin_amdgcn_wmma_f32_16x16x4_f32(
                false, a, false, bmat, (short)0, acc, false, false);
        }
        // Rows 0 and 8 (acc[0] in low/high half) both equal the result: benign
        // same-value store from both half-waves.
        nxt[tile * 16 + n] = gelu_exact(acc[0] + bb[tile * 16 + n]);
        __syncthreads();
        float* t = cur; cur = nxt; nxt = t;
    }

    // Final projection: w12[j] = x . W5[j,:] + b5[j], j < 12 (C=4 x 3)
    if (tid < 12) {
        const float* wr = W5 + tid * D;
        float s = b5[tid];
        for (int k = 0; k < D; ++k) s = fmaf(cur[k], wr[k], s);
        w12_out[tid] = s;
    }
}

// ---------------------------------------------------------------------------
// Kernel 2: bandwidth-bound stream. pred[b,p,d] = sum_c feat[b,c,p]*w[c*3+d].
// Each thread: 4 pixels -> 4x float4 loads (one per channel), 3x float4 stores.
// 7 MB total traffic; this is the 23.3 TB/s floor of the problem.
// ---------------------------------------------------------------------------
__global__ __launch_bounds__(256) void pred_kernel(
    const float* __restrict__ feat, const float* __restrict__ w12,
    float* __restrict__ out)
{
    __shared__ float w[12];
    if (threadIdx.x < 12) w[threadIdx.x] = w12[threadIdx.x];
    __syncthreads();

    const int t = blockIdx.x * 256 + threadIdx.x;   // 0..65535; 4 pixels each
    const int b = t >> 14;                          // batch (q/4 = 16384 per image)
    const int i = t & 16383;                        // float4 index within image

    const float4* fb = (const float4*)(feat + ((size_t)b << 18)); // b*C*H*W
    float4 c0 = fb[i];
    float4 c1 = fb[16384 + i];
    float4 c2 = fb[2 * 16384 + i];
    float4 c3 = fb[3 * 16384 + i];

    float p0[4] = {c0.x, c0.y, c0.z, c0.w};
    float p1[4] = {c1.x, c1.y, c1.z, c1.w};
    float p2[4] = {c2.x, c2.y, c2.z, c2.w};
    float p3[4] = {c3.x, c3.y, c3.z, c3.w};

    float o[12];
#pragma unroll
    for (int j = 0; j < 4; ++j) {
#pragma unroll
        for (int d = 0; d < 3; ++d) {
            o[j * 3 + d] = fmaf(p0[j], w[d],
                           fmaf(p1[j], w[3 + d],
                           fmaf(p2[j], w[6 + d],
                                p3[j] * w[9 + d])));
        }
    }

    // out offset = pixel_base*3 = t*12 floats = t*48 bytes (16B aligned)
    float4* op = (float4*)(out + (size_t)t * 12);
    op[0] = make_float4(o[0], o[1], o[2],  o[3]);
    op[1] = make_float4(o[4], o[5], o[6],  o[7]);
    op[2] = make_float4(o[8], o[9], o[10], o[11]);
}

extern "C" void kernel_launch(void* const* d_in, const int* in_sizes, int n_in,
                              void* d_out, int out_size, void* d_ws, size_t ws_size,
                              hipStream_t stream) {
    const float* feat = (const float*)d_in[0];
    const float* W1   = (const float*)d_in[1];
    const float* b1   = (const float*)d_in[2];
    const float* W2   = (const float*)d_in[3];
    const float* b2   = (const float*)d_in[4];
    const float* W3   = (const float*)d_in[5];
    const float* b3   = (const float*)d_in[6];
    const float* W4   = (const float*)d_in[7];
    const float* b4   = (const float*)d_in[8];
    const float* W5   = (const float*)d_in[9];
    const float* b5   = (const float*)d_in[10];

    float* w12 = (float*)d_ws;          // 12 floats of scratch
    float* out = (float*)d_out;         // (4, 65536, 3) f32

    mlp_weight_kernel<<<1, 512, 0, stream>>>(W1, b1, W2, b2, W3, b3, W4, b4, W5, b5, w12);
    pred_kernel<<<256, 256, 0, stream>>>(feat, w12, out);
}